// PGAgent_81166291960370
// MI455X (gfx1250) — compile-verified
//
#include <hip/hip_runtime.h>
#include <hip/hip_bf16.h>
#include <stdint.h>

// ---- problem constants (from reference) ----
#define B_SZ    1024
#define PAD_LEN 127
#define HID     512
#define NB      128
#define NY      8
#define IN_DIM  640         // 5*(PAD_LEN+1)
#define NA      1024        // NB*NY
#define MAX_TRIP 10
#define MAX_ROW  8

// output layout (floats, concatenated in return order)
#define OUT_PROBS   0
#define OUT_SLOT    (1024*1024)                 // 1,048,576
#define N_SLOT      (B_SZ*NY*MAX_TRIP*MAX_ROW*2)// 1,310,720
#define OUT_LEN     (OUT_SLOT + N_SLOT)
#define N_LW        (N_SLOT/2)                  // 655,360
#define OUT_WID     (OUT_LEN + N_LW)

typedef float v2f __attribute__((ext_vector_type(2)));
typedef float v8f __attribute__((ext_vector_type(8)));

// ---------------------------------------------------------------------------
// Kernel A: valid[b,nb,ny] = any(encoder_mask[b,nb,ny,:]) && !block_mask[b,nb]
// One wave per 4 rows (4*127B = 127 aligned dwords). Coalesced b32 loads,
// per-lane byte->row-flag OR, wave-wide shfl_xor OR-reduce.
// ---------------------------------------------------------------------------
__global__ void yard_valid_kernel(const unsigned char* __restrict__ em,
                                  const unsigned char* __restrict__ bmask,
                                  unsigned char* __restrict__ valid) {
    const int wave = (int)((blockIdx.x * blockDim.x + threadIdx.x) >> 5);
    const int lane = threadIdx.x & 31;
    const int ngroups = (B_SZ * NB * NY) / 4;   // 262,144
    if (wave >= ngroups) return;

    const uint32_t* p = (const uint32_t*)(em + (size_t)wave * 508u);
    uint32_t flags = 0u;
#pragma unroll
    for (int it = 0; it < 4; ++it) {
        int idx = it * 32 + lane;               // dword index within group [0,127)
        if (idx < 127) {
            uint32_t w = p[idx];
            if (w) {
                int bb = idx * 4;               // byte offset within group
#pragma unroll
                for (int k = 0; k < 4; ++k)
                    if ((w >> (8 * k)) & 0xffu)
                        flags |= 1u << ((bb + k) / 127);  // row-in-group 0..3
            }
        }
    }
#pragma unroll
    for (int off = 16; off >= 1; off >>= 1)
        flags |= __shfl_xor(flags, off, 32);

    if (lane < 4) {
        int r = wave * 4 + lane;                // flat (b,nb,ny) row index
        unsigned char any = (unsigned char)((flags >> lane) & 1u);
        valid[r] = (any && !bmask[r >> 3]) ? (unsigned char)1 : (unsigned char)0;
    }
}

// ---------------------------------------------------------------------------
// Kernel B: fused LSTM cell. Each wave computes one 16(batch) x 16(hidden)
// tile of all four gates with V_WMMA_F32_16X16X4_F32 (4 accumulators reuse
// the same A fragment), then applies the sigmoid/tanh epilogue with c0 and
// writes h directly. No divergence -> EXEC all ones for WMMA.
// ---------------------------------------------------------------------------
__global__ __launch_bounds__(128)
void lstm_gates_kernel(const float* __restrict__ x,
                       const float* __restrict__ h0,
                       const float* __restrict__ c0,
                       const float* __restrict__ W_ih,
                       const float* __restrict__ W_hh,
                       const float* __restrict__ b_ih,
                       const float* __restrict__ b_hh,
                       float* __restrict__ h_out) {
    const int wid  = (int)((blockIdx.x * blockDim.x + threadIdx.x) >> 5);
    const int lane = threadIdx.x & 31;
    const int nt   = wid & 31;                  // HID/16 = 32 N tiles
    const int mt   = wid >> 5;                  // B/16   = 64 M tiles
    const int l16  = lane & 15;
    const int mrow = mt * 16 + l16;             // A row for this lane
    const int ncol = nt * 16 + l16;             // hidden column for this lane
    const int kh   = (lane >> 4) << 1;          // K sub-offset 0 or 2

    v8f acc[4] = {};                            // i, f, g, o

    // gates += x @ W_ih^T  (K = IN_DIM)
    for (int kb = 0; kb < IN_DIM; kb += 4) {
        v2f a = *(const v2f*)(x + (size_t)mrow * IN_DIM + kb + kh);
#pragma unroll
        for (int q = 0; q < 4; ++q) {
            v2f b = *(const v2f*)(W_ih + (size_t)(q * HID + ncol) * IN_DIM + kb + kh);
            acc[q] = __builtin_amdgcn_wmma_f32_16x16x4_f32(
                false, a, false, b, (short)0, acc[q], false, false);
        }
    }
    // gates += h0 @ W_hh^T  (K = HID)
    for (int kb = 0; kb < HID; kb += 4) {
        v2f a = *(const v2f*)(h0 + (size_t)mrow * HID + kb + kh);
#pragma unroll
        for (int q = 0; q < 4; ++q) {
            v2f b = *(const v2f*)(W_hh + (size_t)(q * HID + ncol) * HID + kb + kh);
            acc[q] = __builtin_amdgcn_wmma_f32_16x16x4_f32(
                false, a, false, b, (short)0, acc[q], false, false);
        }
    }

    const float bi = b_ih[0 * HID + ncol] + b_hh[0 * HID + ncol];
    const float bf = b_ih[1 * HID + ncol] + b_hh[1 * HID + ncol];
    const float bg = b_ih[2 * HID + ncol] + b_hh[2 * HID + ncol];
    const float bo = b_ih[3 * HID + ncol] + b_hh[3 * HID + ncol];
    const int mhi = (lane >> 4) * 8;            // C/D layout: row = v + 8*(lane/16)

#pragma unroll
    for (int v = 0; v < 8; ++v) {
        const int m = mt * 16 + v + mhi;
        const float gi = acc[0][v] + bi;
        const float gf = acc[1][v] + bf;
        const float gg = acc[2][v] + bg;
        const float go = acc[3][v] + bo;
        const float si = 1.f / (1.f + __expf(-gi));
        const float sf = 1.f / (1.f + __expf(-gf));
        const float so = 1.f / (1.f + __expf(-go));
        const float cc = sf * c0[(size_t)m * HID + ncol] + si * tanhf(gg);
        h_out[(size_t)m * HID + ncol] = so * tanhf(cc);
    }
}

// ---------------------------------------------------------------------------
// Kernel C: logits = h @ W_fc^T + b_fc. Each wave computes a 16(M) x 64(N)
// strip: 4 accumulators over consecutive N tiles share every A fragment
// (4x A reuse, same pipeline shape as the LSTM kernel).
// ---------------------------------------------------------------------------
__global__ __launch_bounds__(128)
void fc_logits_kernel(const float* __restrict__ h,
                      const float* __restrict__ W_fc,
                      const float* __restrict__ b_fc,
                      float* __restrict__ logits) {
    const int wid  = (int)((blockIdx.x * blockDim.x + threadIdx.x) >> 5);
    const int lane = threadIdx.x & 31;
    const int ng   = wid & 15;                  // NA/64 = 16 N groups
    const int mt   = wid >> 4;                  // 64 M tiles
    const int l16  = lane & 15;
    const int mrow = mt * 16 + l16;
    const int nbase = ng * 64;
    const int kh   = (lane >> 4) << 1;

    v8f acc[4] = {};
    for (int kb = 0; kb < HID; kb += 4) {
        v2f a = *(const v2f*)(h + (size_t)mrow * HID + kb + kh);
#pragma unroll
        for (int t = 0; t < 4; ++t) {
            v2f b = *(const v2f*)(W_fc + (size_t)(nbase + t * 16 + l16) * HID + kb + kh);
            acc[t] = __builtin_amdgcn_wmma_f32_16x16x4_f32(
                false, a, false, b, (short)0, acc[t], false, false);
        }
    }

    const int mhi = (lane >> 4) * 8;
#pragma unroll
    for (int t = 0; t < 4; ++t) {
        const int ncol = nbase + t * 16 + l16;
        const float bias = b_fc[ncol];
#pragma unroll
        for (int v = 0; v < 8; ++v) {
            const int m = mt * 16 + v + mhi;
            logits[(size_t)m * NA + ncol] = acc[t][v] + bias;
        }
    }
}

// ---------------------------------------------------------------------------
// Kernel D: masked softmax over NA=1024 columns per row + greedy argmax
// (first occurrence of max, matching jnp.argmax). One 256-thread block/row.
// ---------------------------------------------------------------------------
__global__ __launch_bounds__(256)
void softmax_argmax_kernel(const float* __restrict__ logits,
                           const unsigned char* __restrict__ valid,
                           float* __restrict__ probs,
                           int* __restrict__ action) {
    const int row = blockIdx.x;
    const int t   = threadIdx.x;
    __shared__ float sred[256];
    __shared__ int   sidx[256];

    float ml[4];
    float lmax = -3.4e38f;
    int   lidx = 0;
#pragma unroll
    for (int it = 0; it < 4; ++it) {
        const int c = t + it * 256;             // ascending -> first-max within thread
        const float v = valid[(size_t)row * NA + c]
                        ? logits[(size_t)row * NA + c] : -1e9f;
        ml[it] = v;
        if (v > lmax) { lmax = v; lidx = c; }
    }
    sred[t] = lmax; sidx[t] = lidx;
    __syncthreads();
    for (int s = 128; s > 0; s >>= 1) {
        if (t < s) {
            const float om = sred[t + s]; const int oi = sidx[t + s];
            if (om > sred[t] || (om == sred[t] && oi < sidx[t])) {
                sred[t] = om; sidx[t] = oi;
            }
        }
        __syncthreads();
    }
    const float gmax = sred[0];
    const int   gidx = sidx[0];
    __syncthreads();

    float ls = 0.f;
#pragma unroll
    for (int it = 0; it < 4; ++it) ls += __expf(ml[it] - gmax);
    sred[t] = ls;
    __syncthreads();
    for (int s = 128; s > 0; s >>= 1) {
        if (t < s) sred[t] += sred[t + s];
        __syncthreads();
    }
    const float inv = 1.f / sred[0];

#pragma unroll
    for (int it = 0; it < 4; ++it)
        probs[(size_t)row * NA + t + it * 256] = __expf(ml[it] - gmax) * inv;
    if (t == 0) action[row] = gidx;
}

// ---------------------------------------------------------------------------
// Kernel E: pass-through copies into d_out (tuple outputs concatenated as
// float). barge_slot ints are emitted as float values.
// ---------------------------------------------------------------------------
__global__ void copy_outputs_kernel(const int* __restrict__ bslot,
                                    const float* __restrict__ blen,
                                    const float* __restrict__ bwid,
                                    float* __restrict__ out) {
    const int i = blockIdx.x * blockDim.x + threadIdx.x;
    if (i < N_SLOT) out[OUT_SLOT + i] = (float)bslot[i];
    if (i < N_LW) {
        out[OUT_LEN + i] = blen[i];
        out[OUT_WID + i] = bwid[i];
    }
}

// ---------------------------------------------------------------------------
// Kernel F: per-sample barge-slot scatter update (runs after copy kernel).
// ---------------------------------------------------------------------------
__global__ void barge_update_kernel(const int* __restrict__ action,
                                    const int* __restrict__ barge_count,
                                    const float* __restrict__ block_size,
                                    const float* __restrict__ blen,
                                    const float* __restrict__ bwid,
                                    float* __restrict__ out) {
    const int b = blockIdx.x * blockDim.x + threadIdx.x;
    if (b >= B_SZ) return;
    const int a = action[b];
    const int block_sel = a / NY;
    const int yard      = a % NY;
    const int cur = barge_count[b * NY + yard];
    const float L = block_size[((size_t)b * NB + block_sel) * 5 + 0];
    const float W = block_size[((size_t)b * NB + block_sel) * 5 + 1];
    const size_t base = (((size_t)b * NY + yard) * MAX_TRIP + cur) * MAX_ROW;

    float best = -3.4e38f; int slot = 0;
#pragma unroll
    for (int j = 0; j < MAX_ROW; ++j) {
        const float w = bwid[base + j];
        const float l = blen[base + j];
        const float cand = (w <= 10.0f) ? l : -1.0f;
        if (cand > best) { best = cand; slot = j; }   // first max, as jnp.argmax
    }
    out[OUT_SLOT + (base + slot) * 2 + 1] = (float)block_sel;
    out[OUT_LEN + base + slot] = fmaxf(blen[base + slot], L);
    out[OUT_WID + base + slot] = bwid[base + slot] + W;
}

// ---------------------------------------------------------------------------
extern "C" void kernel_launch(void* const* d_in, const int* in_sizes, int n_in,
                              void* d_out, int out_size, void* d_ws, size_t ws_size,
                              hipStream_t stream) {
    const float* x     = (const float*)d_in[0];
    const float* h0    = (const float*)d_in[1];
    const float* c0    = (const float*)d_in[2];
    const float* W_ih  = (const float*)d_in[3];
    const float* W_hh  = (const float*)d_in[4];
    const float* b_ih  = (const float*)d_in[5];
    const float* b_hh  = (const float*)d_in[6];
    const float* W_fc  = (const float*)d_in[7];
    const float* b_fc  = (const float*)d_in[8];
    const float* block_size = (const float*)d_in[9];
    const float* blen  = (const float*)d_in[10];
    const float* bwid  = (const float*)d_in[11];
    const unsigned char* em = (const unsigned char*)d_in[12];   // bool mask
    const unsigned char* bm = (const unsigned char*)d_in[13];   // bool mask
    const int* bcount  = (const int*)d_in[14];
    const int* bslot   = (const int*)d_in[15];
    float* out = (float*)d_out;

    // workspace: valid (1 MiB) | h (2 MiB) | logits (4 MiB) | action (4 KiB)
    char* ws = (char*)d_ws;
    unsigned char* valid = (unsigned char*)ws;
    float* h      = (float*)(ws + (1u << 20));
    float* logits = (float*)(ws + (1u << 20) + (2u << 20));
    int*   action = (int*)  (ws + (1u << 20) + (2u << 20) + (4u << 20));

    // 262,144 groups, 8 waves/block
    yard_valid_kernel<<<32768, 256, 0, stream>>>(em, bm, valid);
    // 2048 tile-waves, 4 waves/block
    lstm_gates_kernel<<<512, 128, 0, stream>>>(x, h0, c0, W_ih, W_hh, b_ih, b_hh, h);
    // 1024 strip-waves (16M x 64N each), 4 waves/block
    fc_logits_kernel<<<256, 128, 0, stream>>>(h, W_fc, b_fc, logits);
    // one block per batch row
    softmax_argmax_kernel<<<B_SZ, 256, 0, stream>>>(logits, valid, out, action);
    // pass-through copies, then point updates
    copy_outputs_kernel<<<(N_SLOT + 255) / 256, 256, 0, stream>>>(bslot, blen, bwid, out);
    barge_update_kernel<<<(B_SZ + 255) / 256, 256, 0, stream>>>(action, bcount,
                                                                block_size, blen, bwid, out);
}